// NeuralModulatedInput_53644141527174
// MI455X (gfx1250) — compile-verified
//
#include <hip/hip_runtime.h>
#include <hip/hip_bf16.h>

typedef float v2f __attribute__((ext_vector_type(2)));
typedef float v8f __attribute__((ext_vector_type(8)));

#define HIDDEN 64
#define KS 5
#define K2 25
#define CH 8
#define HT 256
#define WD 256
#define PADR 2

__device__ __forceinline__ int reflect_idx(int t) {
    t = (t < 0) ? -t : t;
    t = (t >= HT) ? (2 * HT - 2 - t) : t;
    return t;
}

__global__ __launch_bounds__(32)
void nmi_fused_kernel(const float* __restrict__ input,
                      const float* __restrict__ foa,
                      const float* __restrict__ W1,
                      const float* __restrict__ b1,
                      const float* __restrict__ W2,
                      const float* __restrict__ b2,
                      float* __restrict__ out) {
    __shared__ float kbuf[16][32];   // [pixel][tap], taps padded 25->32

    const int lane = threadIdx.x;            // 0..31
    const int tile = blockIdx.x;              // b * H * (W/16)
    const int xt = tile & 15;                 // W/16 == 16
    const int y  = (tile >> 4) & (HT - 1);
    const int b  = tile >> 12;                // / (16*256)
    const int x0 = xt << 4;

    const int r  = lane & 15;                 // matrix row / column-in-block index
    const int hi = lane >> 4;                 // 0: lanes 0-15, 1: lanes 16-31

    // ---- per-lane pixel coords for the A-matrix row it owns ----
    const float fx = foa[b * 2 + 0];
    const float fy = foa[b * 2 + 1];
    const float dx = (float)(x0 + r) - fx;
    const float dy = (float)y - fy;

    // ---- layer 1 (VALU), produced directly in WMMA A-layout:
    // lane needs h[r][4*kc + 2*hi + t] for kc=0..15, t=0..1
    float hA[32];
#pragma unroll
    for (int kc = 0; kc < 16; ++kc) {
#pragma unroll
        for (int t = 0; t < 2; ++t) {
            const int j = 4 * kc + 2 * hi + t;
            float pre = dx * W1[0 * HIDDEN + j] + dy * W1[1 * HIDDEN + j] + b1[j];
            hA[2 * kc + t] = pre > 0.0f ? pre : 0.0f;
        }
    }

    // ---- W2 into B-layout registers (wave-uniform data, per-lane slices) ----
    float Breg[16][2][2];
#pragma unroll
    for (int kc = 0; kc < 16; ++kc) {
#pragma unroll
        for (int nb = 0; nb < 2; ++nb) {
            const int col  = 16 * nb + r;
            const float mk = (col < K2) ? 1.0f : 0.0f;
            const int colc = (col < K2) ? col : (K2 - 1);
#pragma unroll
            for (int t = 0; t < 2; ++t) {
                const int row = 4 * kc + 2 * hi + t;
                Breg[kc][nb][t] = W2[row * K2 + colc] * mk;
            }
        }
    }

    // ---- C accumulators = b2 broadcast along rows (C/D layout: same col per vgpr) ----
    v8f acc[2];
#pragma unroll
    for (int nb = 0; nb < 2; ++nb) {
        const int col = 16 * nb + r;
        const float cb = (col < K2) ? b2[col] : 0.0f;
#pragma unroll
        for (int v = 0; v < 8; ++v) acc[nb][v] = cb;
    }

    // ---- layer 2: 16 K-chunks x 2 N-blocks of V_WMMA_F32_16X16X4_F32 ----
#pragma unroll
    for (int kc = 0; kc < 16; ++kc) {
        v2f A;
        A[0] = hA[2 * kc];
        A[1] = hA[2 * kc + 1];
#pragma unroll
        for (int nb = 0; nb < 2; ++nb) {
            v2f Bv;
            Bv[0] = Breg[kc][nb][0];
            Bv[1] = Breg[kc][nb][1];
            acc[nb] = __builtin_amdgcn_wmma_f32_16x16x4_f32(
                false, A, false, Bv, (short)0, acc[nb], false, false);
        }
    }

    // ---- scatter D-layout (row = v + 8*hi, tap = 16*nb + r) into LDS ----
#pragma unroll
    for (int nb = 0; nb < 2; ++nb) {
#pragma unroll
        for (int v = 0; v < 8; ++v) {
            kbuf[v + 8 * hi][16 * nb + r] = acc[nb][v];
        }
    }
    __syncthreads();

    // ---- modulation: 2 lanes per pixel, 4 channels each ----
    const int p = r;
    const int x = x0 + p;
    const int c0 = hi * 4;
#pragma unroll
    for (int cc = 0; cc < 4; ++cc) {
        const int c = c0 + cc;
        const float* in_bc = input + (size_t)(b * CH + c) * HT * WD;
        float s = 0.0f;
        int idx = 0;
#pragma unroll
        for (int ky = 0; ky < KS; ++ky) {
            const int yy = reflect_idx(y + ky - PADR);
            const float* rowp = in_bc + (size_t)yy * WD;
#pragma unroll
            for (int kx = 0; kx < KS; ++kx) {
                const int xx = reflect_idx(x + kx - PADR);
                s += kbuf[p][idx] * rowp[xx];
                ++idx;
            }
        }
        out[((size_t)(b * CH + c) * HT + y) * WD + x] = s;
    }
}

extern "C" void kernel_launch(void* const* d_in, const int* in_sizes, int n_in,
                              void* d_out, int out_size, void* d_ws, size_t ws_size,
                              hipStream_t stream) {
    const float* input = (const float*)d_in[0];  // (8,8,256,256)
    const float* foa   = (const float*)d_in[1];  // (8,2)
    const float* W1    = (const float*)d_in[2];  // (2,64)
    const float* b1    = (const float*)d_in[3];  // (64,)
    const float* W2    = (const float*)d_in[4];  // (64,25)
    const float* b2    = (const float*)d_in[5];  // (25,)
    float* out = (float*)d_out;

    const int nblocks = 8 * 256 * (256 / 16);    // 32768 single-wave tiles
    nmi_fused_kernel<<<dim3(nblocks), dim3(32), 0, stream>>>(
        input, foa, W1, b1, W2, b2, out);
}